// BitKHopSampler_81174881894713
// MI455X (gfx1250) — compile-verified
//
#include <hip/hip_runtime.h>

typedef __attribute__((ext_vector_type(4))) float f32x4;
typedef __attribute__((ext_vector_type(4))) int   i32x4;
// GCC-style vector type matching the builtin's expected pointee (int __vector(4)).
typedef int v4i __attribute__((vector_size(16)));

#define BATCH        512
#define NUM_VARS     1024
#define NUM_SAMPLES  128
#define NUM_HOPS     4
#define THREADS      256   // 8 wave32 waves
#define S_PER_BLOCK  16    // samples per block

// CDNA5 async global->LDS path (ASYNCcnt), guarded so compile never breaks.
#if __has_builtin(__builtin_amdgcn_global_load_async_to_lds_b128) && \
    __has_builtin(__builtin_amdgcn_s_wait_asynccnt)
#define HAVE_ASYNC_LDS 1
#else
#define HAVE_ASYNC_LDS 0
#endif

__global__ __launch_bounds__(THREADS, 2) void BitKHopSampler_kernel(
    const float* __restrict__ y,
    const int*   __restrict__ idx,
    float*       __restrict__ out)
{
    const int t  = threadIdx.x;          // float4 slot within the 1024-wide row
    const int b  = blockIdx.y;           // batch
    const int s0 = blockIdx.x * S_PER_BLOCK;

    const float* ysrc = y + (size_t)b * NUM_VARS;

#if HAVE_ASYNC_LDS
    // Stage the 4 KB y-row via the async global->LDS DMA path:
    // 256 threads x 16 B = one GLOBAL_LOAD_ASYNC_TO_LDS_B128 per thread.
    __shared__ float yrow[NUM_VARS];
    __builtin_amdgcn_global_load_async_to_lds_b128(
        (__attribute__((address_space(1))) v4i*)(ysrc + 4 * t),
        (__attribute__((address_space(3))) v4i*)(yrow + 4 * t),
        /*offset=*/0, /*cpol=*/0);
    __builtin_amdgcn_s_wait_asynccnt(0);
    __syncthreads();
    const f32x4 yv = *(const f32x4*)(yrow + 4 * t);
#else
    // Fallback: direct b128 load; y is tiny (2 MB) and L2-resident.
    const f32x4 yv = *(const f32x4*)(ysrc + 4 * t);
#endif

    const f32x4 fv = 1.0f - yv;          // flipped values, computed once

    // Wave-uniform index pointer -> compiler can use s_load_b128 per sample.
    const i32x4* ip   = (const i32x4*)(idx + ((size_t)b * NUM_SAMPLES + s0) * NUM_HOPS);
    float*       orow = out + ((size_t)b * NUM_SAMPLES + s0) * NUM_VARS;
    const int    v0   = t * 4;

#pragma unroll
    for (int s = 0; s < S_PER_BLOCK; ++s) {
        const i32x4 iv = ip[s];
        f32x4 o;
#pragma unroll
        for (int j = 0; j < 4; ++j) {
            const int  v   = v0 + j;
            const bool hit = (v == iv.x) | (v == iv.y) | (v == iv.z) | (v == iv.w);
            o[j] = hit ? fv[j] : yv[j];
        }
        // Output (256 MB) exceeds the 192 MB L2 and is write-once:
        // stream it with a non-temporal b128 store so y stays cached.
        __builtin_nontemporal_store(o, (f32x4*)(orow + (size_t)s * NUM_VARS) + t);
    }
}

extern "C" void kernel_launch(void* const* d_in, const int* in_sizes, int n_in,
                              void* d_out, int out_size, void* d_ws, size_t ws_size,
                              hipStream_t stream) {
    // setup_inputs order: a(0), b(1), c(2), y(3), idx(4). a/b/c are unused by
    // the reference output.
    const float* y   = (const float*)d_in[3];
    const int*   idx = (const int*)d_in[4];
    float*       out = (float*)d_out;

    dim3 grid(NUM_SAMPLES / S_PER_BLOCK, BATCH);   // 8 x 512 = 4096 blocks
    dim3 block(THREADS);
    BitKHopSampler_kernel<<<grid, block, 0, stream>>>(y, idx, out);
}